// DotProductAttention_31877247271037
// MI455X (gfx1250) — compile-verified
//
#include <hip/hip_runtime.h>

// Fused single-pass DotProductAttention for MI455X (gfx1250, wave32).
// x: (B=2048, T=512, D=256) f32 -> out = [context (B,D) | weights (B,T)] f32.
// Online-softmax fusion: one HBM pass over the 1 GiB tensor (~46 us floor at
// 23.3 TB/s) instead of two. Per wave: async-copy a 16t x 256d tile (16 KB,
// contiguous) into LDS via GLOBAL_LOAD_ASYNC_TO_LDS_B128, compute the tile's
// scores with V_WMMA_F32_16X16X4_F32 from LDS, update running (max,sum) with
// rescaling, and accumulate exp-weighted context from the same LDS tile.

typedef float v2f __attribute__((ext_vector_type(2)));
typedef float v8f __attribute__((ext_vector_type(8)));

#define TT 512
#define DD 256
#define NWAVES 8
#define TILE_T 16

#if defined(__gfx1250__) && __has_builtin(__builtin_amdgcn_global_load_async_to_lds_b128)
#define ASYNC_LDS 1
// Builtin signature (from hipcc diagnostic): param0 = global (AS1, prints as
// __device__) pointer to <4 x i32>, non-const; param1 = LDS side; then two
// immediate ints (offset, cpol).
typedef __attribute__((__vector_size__(4 * sizeof(int)))) int v4i_t;
typedef __attribute__((address_space(1))) v4i_t* g4p_t;
typedef __attribute__((address_space(3))) v4i_t* l4p_t;
#else
#define ASYNC_LDS 0
#endif

static __device__ __forceinline__ void tile_copy_b128(const float* g, float* l) {
#if ASYNC_LDS
  // async DMA: global -> LDS, 16B per lane, tracked by ASYNCcnt
  __builtin_amdgcn_global_load_async_to_lds_b128((g4p_t)g, (l4p_t)l, 0, 0);
#else
  *reinterpret_cast<float4*>(l) = *reinterpret_cast<const float4*>(g);
#endif
}

__global__ __launch_bounds__(256) void attn_fused_kernel(
    const float* __restrict__ x, float* __restrict__ ctx_out,
    float* __restrict__ w_out)
{
  const int b    = blockIdx.x;
  const int tid  = threadIdx.x;
  const int wave = tid >> 5;
  const int lane = tid & 31;
  const float* xb = x + (size_t)b * (TT * DD);

  __shared__ float tileAll[NWAVES * TILE_T * DD]; // 128 KB: one 16KB tile/wave
  __shared__ float qs[DD];                        // query row
  __shared__ float sc[TT];                        // raw scores
  __shared__ float wm[NWAVES], wl[NWAVES];        // per-wave online state

  qs[tid] = xb[(size_t)(TT - 1) * DD + tid];
  __syncthreads();

  // WMMA f32 16x16x4 fragment mapping (ISA 7.12.2):
  // A: lanes 0-15 = (row=lane, K=0..1), lanes 16-31 = (row=lane-16, K=2..3)
  // B: all 16 columns carry the query value for that K.
  const int row_in_tile = lane & 15;
  const int khalf       = lane >> 4;

  float* lt         = tileAll + wave * (TILE_T * DD);
  const float* frow = lt + row_in_tile * DD + khalf * 2;
  const float* crow = lt + lane * 8;      // this lane's d-slice for context

  float m_w = -__builtin_inff();
  float l_w = 0.0f;
  float ctx[8] = {0.f, 0.f, 0.f, 0.f, 0.f, 0.f, 0.f, 0.f};

  for (int j = 0; j < 4; ++j) {                   // 8 waves x 4 tiles = 32
    const int t0 = TILE_T * (wave + NWAVES * j);
    const float* gt = xb + (size_t)t0 * DD;       // contiguous 16 KB tile

#if ASYNC_LDS
    // async LDS writes are NOT ordered vs prior same-wave DS reads:
    // drain DScnt before overwriting the tile buffer.
    asm volatile("s_wait_dscnt 0x0" ::: "memory");
#endif
    #pragma unroll
    for (int i = 0; i < 32; ++i)                  // 32 x (32 lanes x 16B)
      tile_copy_b128(gt + i * 128 + lane * 4, lt + i * 128 + lane * 4);
#if ASYNC_LDS
    asm volatile("s_wait_asynccnt 0x0" ::: "memory");
#endif

    // ---- tile scores via WMMA (replicated across the 16 B-columns) ----
    v8f c = {};
    #pragma unroll 4
    for (int k0 = 0; k0 < DD; k0 += 4) {
      v2f a = *reinterpret_cast<const v2f*>(frow + k0);   // ds_load_b64
      v2f bq;
      bq.x = qs[k0 + khalf];
      bq.y = qs[k0 + khalf + 2];
      c = __builtin_amdgcn_wmma_f32_16x16x4_f32(
          false, a, false, bq, (short)0, c, false, false);
    }
    // C/D: vgpr r -> M=r (lanes 0-15) / M=8+r (lanes 16-31); take column 0.
    if (lane == 0) {
      #pragma unroll
      for (int r = 0; r < 8; ++r) sc[t0 + r] = c[r];
    }
    if (lane == 16) {
      #pragma unroll
      for (int r = 0; r < 8; ++r) sc[t0 + 8 + r] = c[r];
    }
    // same-wave LDS ops are in-order: safe to read sc back without barrier

    // ---- online softmax update + context accumulation from LDS tile ----
    float e[TILE_T];
    float snew = m_w;
    #pragma unroll
    for (int r = 0; r < TILE_T; ++r) {
      e[r] = sc[t0 + r];
      snew = fmaxf(snew, e[r]);
    }
    const float scale = __expf(m_w - snew);       // exp(-inf)=0 on first tile
    float esum = 0.f;
    #pragma unroll
    for (int r = 0; r < TILE_T; ++r) {
      e[r] = __expf(e[r] - snew);
      esum += e[r];
    }
    l_w = fmaf(l_w, scale, esum);
    m_w = snew;
    #pragma unroll
    for (int d = 0; d < 8; ++d) ctx[d] *= scale;
    #pragma unroll
    for (int r = 0; r < TILE_T; ++r) {
      const float wt = e[r];
      #pragma unroll
      for (int d = 0; d < 8; ++d)
        ctx[d] = fmaf(wt, crow[r * DD + d], ctx[d]);
    }
  }

  // ---- merge per-wave online-softmax partials across the 8 waves ----
  if (lane == 0) { wm[wave] = m_w; wl[wave] = l_w; }
  __syncthreads();                                // also: all tile reads done

  float M = wm[0];
  #pragma unroll
  for (int w = 1; w < NWAVES; ++w) M = fmaxf(M, wm[w]);
  float L = 0.f;
  #pragma unroll
  for (int w = 0; w < NWAVES; ++w) L += wl[w] * __expf(wm[w] - M);

  float* cred = tileAll;                          // reuse tile LDS (free now)
  const float ms = __expf(m_w - M);
  #pragma unroll
  for (int d = 0; d < 8; ++d)
    cred[wave * DD + lane * 8 + d] = ctx[d] * ms;
  __syncthreads();

  const float invL = 1.0f / L;
  float acc = 0.f;
  #pragma unroll
  for (int w = 0; w < NWAVES; ++w) acc += cred[w * DD + tid];
  ctx_out[(size_t)b * DD + tid] = acc * invL;

  w_out[(size_t)b * TT + tid]       = __expf(sc[tid] - M) * invL;
  w_out[(size_t)b * TT + tid + 256] = __expf(sc[tid + 256] - M) * invL;
}

extern "C" void kernel_launch(void* const* d_in, const int* in_sizes, int n_in,
                              void* d_out, int out_size, void* d_ws, size_t ws_size,
                              hipStream_t stream) {
  const float* x = (const float*)d_in[0];
  const int Bn = in_sizes[0] / (TT * DD);         // 2048
  float* ctx = (float*)d_out;                     // (B, D)
  float* wgt = (float*)d_out + (size_t)Bn * DD;   // (B, T, 1) flat

  attn_fused_kernel<<<Bn, 256, 0, stream>>>(x, ctx, wgt);
}